// MyRotaryPositionalEmbedding_21406117004144
// MI455X (gfx1250) — compile-verified
//
#include <hip/hip_runtime.h>

// ---------------------------------------------------------------------------
// RoPE for MI455X (gfx1250, wave32).
// Memory-bound: 64MiB in + 64MiB out + 0.5MiB positions @ 23.3 TB/s ~= 5.5us.
// Strategy:
//   1) one-time 4MiB lane-layout cos/sin table (L2-resident; 192MB L2)
//   2) streaming apply: one wave32 == one 128-dim token row (32 lanes x 16B),
//      x-stream staged via gfx1250 async global->LDS DMA
//      (global_load_async_to_lds_b128 + s_wait_asynccnt), double-buffered,
//      non-temporal b128 stores for the write-once output stream.
// ---------------------------------------------------------------------------

typedef float v4f __attribute__((ext_vector_type(4)));

#define ROPE_DK        128
#define ROPE_LANES     32            // 128 f32 dims / 4 per lane -> one wave32 per token
#define ROPE_MAXSEQ    8192
#define ROPE_TAB_V4    (ROPE_MAXSEQ * ROPE_LANES)   // 262144 float4 = 4 MiB

// log2(10000) / 64 : inv_freq(k) = 2^(-k * log2(theta)/64), k = 0..63
#define ROPE_L2T_OVER64 (13.287712379549449f / 64.0f)

// Issue per-lane async DMA of 16B from global into this lane's LDS slot.
// GV addressing mode: vdst = LDS byte address, vaddr = 64-bit global address.
__device__ __forceinline__ void rope_async_ld_b128(unsigned lds_addr, const v4f* gptr) {
    asm volatile("global_load_async_to_lds_b128 %0, %1, off"
                 :: "v"(lds_addr), "v"(gptr)
                 : "memory");
}

// ---------------------------------------------------------------------------
// Kernel 1: build per-(pos, lane) table: float4{cos(2l), sin(2l), cos(2l+1),
// sin(2l+1)}. 262144 threads, 2 accurate sincosf each -> negligible one-time
// cost vs the 6us stream (angles reach ~8191 rad, need real range reduction).
// ---------------------------------------------------------------------------
__global__ __launch_bounds__(256) void rope_build_table(v4f* __restrict__ tab) {
    const int tid = blockIdx.x * blockDim.x + threadIdx.x;   // exact-sized grid
    const int pos = tid >> 5;        // 0..8191
    const int l   = tid & 31;        // lane-within-token
    const float p = (float)pos;
    const float f0 = exp2f(-(float)(2 * l)     * ROPE_L2T_OVER64);
    const float f1 = exp2f(-(float)(2 * l + 1) * ROPE_L2T_OVER64);
    float s0, c0, s1, c1;
    sincosf(p * f0, &s0, &c0);
    sincosf(p * f1, &s1, &c1);
    v4f e; e.x = c0; e.y = s0; e.z = c1; e.w = s1;
    tab[tid] = e;
}

// ---------------------------------------------------------------------------
// Kernel 2: streaming apply via async global->LDS double-buffer.
// i indexes float4 chunks of x: token t = i>>5, lane l = i&31.
// Each thread owns one 16B slot per buffer; ASYNCcnt is per-wave and async
// loads complete in order, so no cross-wave barrier is needed.
// ---------------------------------------------------------------------------
__global__ __launch_bounds__(256) void rope_apply_async(const v4f* __restrict__ x,
                                                        const int* __restrict__ tpos,
                                                        const v4f* __restrict__ tab,
                                                        v4f* __restrict__ out,
                                                        int n_vec) {
    __shared__ v4f buf[2][256];      // 8 KiB double buffer, per-thread 16B slots
    const int tid = threadIdx.x;
    // Low 32 bits of a generic pointer into LDS == DS-space byte address
    // (local->generic addrspacecast places the LDS offset in addr[31:0]).
    const unsigned lds0 = (unsigned)(size_t)&buf[0][tid];
    const unsigned lds1 = (unsigned)(size_t)&buf[1][tid];

    const int stride = gridDim.x * blockDim.x;
    int i = blockIdx.x * blockDim.x + tid;
    if (i >= n_vec) return;

    rope_async_ld_b128(lds0, &x[i]);             // prologue: stage chunk 0
    int cur = 0;
    while (true) {
        const int  inext = i + stride;
        const bool more  = inext < n_vec;
        if (more) {
            rope_async_ld_b128(cur ? lds0 : lds1, &x[inext]);   // stage next
            asm volatile("s_wait_asynccnt 0x1" ::: "memory");   // current done
        } else {
            asm volatile("s_wait_asynccnt 0x0" ::: "memory");   // drain
        }

        const v4f xv = buf[cur][tid];            // ds_load_b128 from own slot
        const int t  = i >> 5;
        const int l  = i & 31;
        const int p  = tpos[t];                  // wave-broadcast, coalesced
        const v4f cs = tab[(p << 5) + l];        // {c0,s0,c1,s1}, L2-resident

        v4f y;
        y.x = cs.x * xv.x - cs.y * xv.y;
        y.y = cs.y * xv.x + cs.x * xv.y;
        y.z = cs.z * xv.z - cs.w * xv.w;
        y.w = cs.w * xv.z + cs.z * xv.w;
        __builtin_nontemporal_store(y, &out[i]); // write-once stream, keep L2 for tab

        if (!more) break;
        i = inext;
        cur ^= 1;
    }
}

// ---------------------------------------------------------------------------
// Fallback (only if workspace < 4MiB): compute sincos inline. Same mapping.
// ---------------------------------------------------------------------------
__global__ __launch_bounds__(256) void rope_direct(const v4f* __restrict__ x,
                                                   const int* __restrict__ tpos,
                                                   v4f* __restrict__ out,
                                                   int n_vec) {
    const int stride = gridDim.x * blockDim.x;
    int i = blockIdx.x * blockDim.x + threadIdx.x;
    for (; i < n_vec; i += stride) {
        if (i + stride < n_vec) {
            __builtin_prefetch(&x[i + stride], 0, 0);   // -> global_prefetch_b8
        }
        const int t = i >> 5;
        const int l = i & 31;
        const v4f xv = __builtin_nontemporal_load(&x[i]);
        const float p = (float)tpos[t];
        const float f0 = exp2f(-(float)(2 * l)     * ROPE_L2T_OVER64);
        const float f1 = exp2f(-(float)(2 * l + 1) * ROPE_L2T_OVER64);
        float s0, c0, s1, c1;
        sincosf(p * f0, &s0, &c0);
        sincosf(p * f1, &s1, &c1);
        v4f y;
        y.x = c0 * xv.x - s0 * xv.y;
        y.y = s0 * xv.x + c0 * xv.y;
        y.z = c1 * xv.z - s1 * xv.w;
        y.w = s1 * xv.z + c1 * xv.w;
        __builtin_nontemporal_store(y, &out[i]);
    }
}

extern "C" void kernel_launch(void* const* d_in, const int* in_sizes, int n_in,
                              void* d_out, int out_size, void* d_ws, size_t ws_size,
                              hipStream_t stream) {
    const v4f* x    = (const v4f*)d_in[0];   // [16, 8192, 128] f32, viewed as float4
    const int* tpos = (const int*)d_in[1];   // [16, 8192] i32
    v4f*       out  = (v4f*)d_out;

    const int n     = in_sizes[0];           // 16*8192*128 = 16,777,216
    const int n_vec = n >> 2;                // 4,194,304 float4 chunks

    const size_t tab_bytes = (size_t)ROPE_TAB_V4 * sizeof(v4f);  // 4 MiB

    if (ws_size >= tab_bytes) {
        v4f* tab = (v4f*)d_ws;
        // exact grid: 8192 * 32 / 256 = 1024 blocks
        rope_build_table<<<ROPE_TAB_V4 / 256, 256, 0, stream>>>(tab);
        // 4096 blocks x 256 threads: 4 float4 per thread, 32K waves -> HBM-saturating
        rope_apply_async<<<4096, 256, 0, stream>>>(x, tpos, tab, out, n_vec);
    } else {
        rope_direct<<<8192, 256, 0, stream>>>(x, tpos, out, n_vec);
    }
}